// PositionAttention_87428354277846
// MI455X (gfx1250) — compile-verified
//
#include <hip/hip_runtime.h>

// PositionAttention fused flash-attention for gfx1250 (MI455X).
// B=8, C=128, N=4096. attention = softmax(rowmax - energy) == softmax(-energy).
// v2: Tensor Data Mover (TDM) double-buffered K/V tile loads, c-major LDS tiles.

typedef __attribute__((ext_vector_type(2))) float v2f;
typedef __attribute__((ext_vector_type(8))) float v8f;
typedef __attribute__((ext_vector_type(4))) unsigned int tdm_v4u;
typedef __attribute__((ext_vector_type(8))) int          tdm_v8i;
typedef __attribute__((ext_vector_type(4))) int          tdm_v4i;

#define BB 8
#define CC 128
#define NN 4096
#define QT 64   // query rows per workgroup
#define KT 64   // key rows per inner block

__device__ __forceinline__ float redmax16(float v) {
  v = fmaxf(v, __shfl_xor(v, 1, 32));
  v = fmaxf(v, __shfl_xor(v, 2, 32));
  v = fmaxf(v, __shfl_xor(v, 4, 32));
  v = fmaxf(v, __shfl_xor(v, 8, 32));
  return v;
}
__device__ __forceinline__ float redsum16(float v) {
  v += __shfl_xor(v, 1, 32);
  v += __shfl_xor(v, 2, 32);
  v += __shfl_xor(v, 4, 32);
  v += __shfl_xor(v, 8, 32);
  return v;
}

// TDM: DMA a 2D tile [CC rows x tile_w cols] of f32 from the row-major tensor
// x_b[CC][NN] (row stride NN elements) into LDS. pad_enable inserts 1 DWORD
// after every tile_w DWORDs -> LDS rows of (tile_w+1) floats, bank-friendly.
// tile_w must equal 2<<pad_code (here 64 -> code 5).
__device__ __forceinline__ void tdm_load_tile_f32(unsigned lds_off,
                                                  const float* g,
                                                  unsigned pad_code,
                                                  unsigned tile_w) {
  unsigned long long ga = (unsigned long long)(uintptr_t)g;
  tdm_v4u g0;
  g0.x = 1u;                                               // count=1, user D#
  g0.y = lds_off;                                          // LDS byte address
  g0.z = (unsigned)ga;                                     // global addr lo
  g0.w = (unsigned)((ga >> 32) & 0x1FFFFFFu) | (2u << 30); // addr[56:32]|type=2
  tdm_v8i g1;
  g1[0] = (int)((2u << 16)                // data_size = 4 bytes
              | (1u << 20)                // pad_enable
              | (pad_code << 22));        // pad_interval; pad_amount=0 (1 DWORD)
  g1[1] = (int)((NN & 0xFFFFu) << 16);                       // tensor_dim0 lo16
  g1[2] = (int)((NN >> 16) & 0xFFFFu) | (int)((CC & 0xFFFFu) << 16); // d0hi|d1lo
  g1[3] = (int)(tile_w << 16);                               // d1 hi=0 | tile_dim0
  g1[4] = (int)CC;                                           // tile_dim1 | tile_dim2=0
  g1[5] = (int)NN;                                           // tensor_dim0_stride lo32
  g1[6] = 0;
  g1[7] = 0;
  tdm_v4i z4 = {0, 0, 0, 0};
#if __clang_major__ >= 23
  tdm_v8i z8 = {0, 0, 0, 0, 0, 0, 0, 0};
  __builtin_amdgcn_tensor_load_to_lds(g0, g1, z4, z4, z8, 0);
#else
  __builtin_amdgcn_tensor_load_to_lds(g0, g1, z4, z4, 0);
#endif
}

__global__ __launch_bounds__(128, 1)
void pa_flash_tdm_kernel(const float* __restrict__ x,
                         const float* __restrict__ gammap,
                         float* __restrict__ out) {
  // c-major tiles (match TDM's natural copy order), 65-float rows (TDM pad).
  __shared__ float Qc[CC][65];       // Qc[c][i] = x[b][c][n0+i]
  __shared__ float Kc[2][CC][65];    // Kc[buf][c][m] = x[b][c][kb*KT+m]
  __shared__ float Ps[4][16][66];    // per-wave P strip (D-layout -> A-layout)

  const int tid  = threadIdx.x;
  const int wave = tid >> 5;
  const int lane = tid & 31;
  const int n16  = lane & 15;
  const int h    = lane >> 4;
  const int qrow = wave * 16;
  const int b    = blockIdx.y;
  const int n0   = blockIdx.x * QT;

  const float* xb = x + (size_t)b * CC * NN;

  // Pre-issue Q tile and first K tile on the TDM engine (wave 0 only;
  // TDM ignores EXEC, so gate with a wave-uniform branch).
  if (wave == 0) {
    tdm_load_tile_f32((unsigned)(size_t)&Qc[0][0],    xb + n0, 5, QT);
    tdm_load_tile_f32((unsigned)(size_t)&Kc[0][0][0], xb,      5, KT);
  }

  v8f O[8] = {};                 // O strip [16 x 128] in D-fragment layout
  float mstate[8], lstate[8];    // running max / sum for rows r + 8*h
#pragma unroll
  for (int r = 0; r < 8; ++r) { mstate[r] = -1.0e30f; lstate[r] = 0.0f; }

  for (int kb = 0; kb < NN / KT; ++kb) {
    const int cur = kb & 1;
    if (wave == 0) {
      if (kb + 1 < NN / KT) {
        // prefetch next K tile into the idle buffer (consumed last iteration,
        // end-of-iteration barrier guarantees all waves finished reading it)
        tdm_load_tile_f32((unsigned)(size_t)&Kc[cur ^ 1][0][0],
                          xb + (size_t)(kb + 1) * KT, 5, KT);
        __builtin_amdgcn_s_wait_tensorcnt(1);   // current tile landed
      } else {
        __builtin_amdgcn_s_wait_tensorcnt(0);
      }
    }
    __syncthreads();

    float (*K)[65] = Kc[cur];

    // ---- S = Q * K^T : 4 tiles of 16x16, K-dim = 128 in steps of 4 ----
    v8f S[4] = {};
#pragma unroll 4
    for (int kk = 0; kk < 32; ++kk) {
      int k2 = 4 * kk + 2 * h;
      v2f a;
      a.x = Qc[k2][qrow + n16];          // A[i][c] = Qc[c][qrow+i]
      a.y = Qc[k2 + 1][qrow + n16];
#pragma unroll
      for (int jt = 0; jt < 4; ++jt) {
        v2f bv;
        bv.x = K[k2][jt * 16 + n16];     // B[k=c][n=j] = Kc[c][j]
        bv.y = K[k2 + 1][jt * 16 + n16];
        S[jt] = __builtin_amdgcn_wmma_f32_16x16x4_f32(
            false, a, false, bv, (short)0, S[jt], false, false);
      }
    }

    // ---- online softmax on scores = -S ----
#pragma unroll
    for (int r = 0; r < 8; ++r) {
      float cm = -S[0][r];
#pragma unroll
      for (int jt = 1; jt < 4; ++jt) cm = fmaxf(cm, -S[jt][r]);
      cm = redmax16(cm);
      float mn = fmaxf(mstate[r], cm);
      float al = __expf(mstate[r] - mn);
      float rs = 0.0f;
#pragma unroll
      for (int jt = 0; jt < 4; ++jt) {
        float pv = __expf(-S[jt][r] - mn);
        Ps[wave][r + 8 * h][jt * 16 + n16] = pv;
        rs += pv;
      }
      rs = redsum16(rs);
      lstate[r] = lstate[r] * al + rs;
      mstate[r] = mn;
#pragma unroll
      for (int ct = 0; ct < 8; ++ct) O[ct][r] *= al;
    }

    // ---- O += P * V : 8 tiles of 16x16, K-dim = 64 in steps of 4 ----
#pragma unroll 4
    for (int kk = 0; kk < 16; ++kk) {
      int k2 = 4 * kk + 2 * h;
      v2f a;
      a.x = Ps[wave][n16][k2];
      a.y = Ps[wave][n16][k2 + 1];
#pragma unroll
      for (int ct = 0; ct < 8; ++ct) {
        v2f bv;
        bv.x = K[ct * 16 + n16][k2];     // B[k=j][n=c] = Kc[c][j]
        bv.y = K[ct * 16 + n16][k2 + 1];
        O[ct] = __builtin_amdgcn_wmma_f32_16x16x4_f32(
            false, a, false, bv, (short)0, O[ct], false, false);
      }
    }
    __syncthreads();   // all waves done reading Kc[cur] before it is refilled
  }

  // ---- epilogue: normalize, stage through LDS, coalesced gamma*O + x ----
  float (*Os)[65] = Kc[0];               // both K buffers are dead now
#pragma unroll
  for (int r = 0; r < 8; ++r) {
    float inv = 1.0f / lstate[r];
    int i = qrow + r + 8 * h;
#pragma unroll
    for (int ct = 0; ct < 8; ++ct)
      Os[ct * 16 + n16][i] = O[ct][r] * inv;   // Os[c][i]
  }
  __syncthreads();

  const float g = gammap[0];
  float* ob = out + (size_t)b * CC * NN;
  for (int idx = tid; idx < QT * CC; idx += 128) {
    int i = idx & (QT - 1);
    int c = idx >> 6;
    size_t off = (size_t)c * NN + n0 + i;
    ob[off] = g * Os[c][i] + xb[off];
  }
}

extern "C" void kernel_launch(void* const* d_in, const int* in_sizes, int n_in,
                              void* d_out, int out_size, void* d_ws, size_t ws_size,
                              hipStream_t stream) {
  (void)in_sizes; (void)n_in; (void)out_size; (void)d_ws; (void)ws_size;
  const float* points = (const float*)d_in[0];
  const float* gamma  = (const float*)d_in[1];
  float* out = (float*)d_out;

  dim3 grid(NN / QT, BB);   // 64 query tiles x 8 batches = 512 workgroups
  pa_flash_tdm_kernel<<<grid, 128, 0, stream>>>(points, gamma, out);
}